// SAGPoolNet_53412213293450
// MI455X (gfx1250) — compile-verified
//
#include <hip/hip_runtime.h>

// Problem constants (match reference)
#define BGRAPHS 512
#define NPG     128
#define EPG     2048
#define INF_    128
#define HID     128
#define KEEPK   64
#define NTOTAL  (BGRAPHS * NPG)

typedef __attribute__((ext_vector_type(2))) float v2f;
typedef __attribute__((ext_vector_type(8))) float v8f;
typedef int v4i_ __attribute__((vector_size(16)));   // matches builtin param

// --------------------------------------------------------------------------
// Async global->LDS staging (GLOBAL_LOAD_ASYNC_TO_LDS_B128, ASYNCcnt), with
// a guarded fallback to plain load/store staging so compilation never breaks.
// Builtin signature (from hipcc diagnostic): (v4i AS1*, v4i AS3*, Ii, Ii).
// --------------------------------------------------------------------------
#if defined(__has_builtin)
#if __has_builtin(__builtin_amdgcn_global_load_async_to_lds_b128)
#define USE_ASYNC_LDS 1
#endif
#endif

__device__ __forceinline__ void stage_b128(const void* g, void* l) {
#ifdef USE_ASYNC_LDS
  __builtin_amdgcn_global_load_async_to_lds_b128(
      (__attribute__((address_space(1))) v4i_*)g,
      (__attribute__((address_space(3))) v4i_*)l, 0, 0);
#else
  *reinterpret_cast<float4*>(l) = *reinterpret_cast<const float4*>(g);
#endif
}

__device__ __forceinline__ void stage_wait() {
#ifdef USE_ASYNC_LDS
#if __has_builtin(__builtin_amdgcn_s_wait_asynccnt)
  __builtin_amdgcn_s_wait_asynccnt(0);
#else
  asm volatile("s_wait_asynccnt 0" ::: "memory");
#endif
#endif
}

// --------------------------------------------------------------------------
// fp32 GEMM C[M,128] = A[M,K] @ B[K,128] (+bias) via V_WMMA_F32_16X16X4_F32.
// One wave per 16x128 output strip: A fragment loaded once per K-step and
// reused across all 8 N-tiles (8 accumulators). N must be 128.
// Grid sized exactly -> EXEC all-ones in every wave (WMMA requirement).
// --------------------------------------------------------------------------
__global__ __launch_bounds__(256) void gemm_f32_wmma(
    const float* __restrict__ A, const float* __restrict__ B,
    const float* __restrict__ bias, float* __restrict__ C,
    int M, int N, int K) {
  const int wave = (blockIdx.x * blockDim.x + threadIdx.x) >> 5;
  const int lane = threadIdx.x & 31;
  const int tm   = wave << 4;
  if (tm >= M) return;  // wave-uniform

  const int half = lane >> 4;   // 0: lanes 0-15, 1: lanes 16-31
  const int l    = lane & 15;

  v8f acc[8];
#pragma unroll
  for (int i = 0; i < 8; ++i) acc[i] = (v8f){};

  const float* arow = A + (size_t)(tm + l) * K;
  for (int k0 = 0; k0 < K; k0 += 4) {
    const int ka = k0 + (half << 1);
    // A 16x4 layout: VGPR0 = {K=k0 | K=k0+2}, VGPR1 = {K=k0+1 | K=k0+3}
    v2f a;
    a.x = arow[ka];
    a.y = arow[ka + 1];
    const float* brow0 = B + (size_t)ka * N + l;
    const float* brow1 = brow0 + N;
#pragma unroll
    for (int tn8 = 0; tn8 < 8; ++tn8) {
      v2f b;
      b.x = brow0[tn8 * 16];
      b.y = brow1[tn8 * 16];
      acc[tn8] = __builtin_amdgcn_wmma_f32_16x16x4_f32(
          false, a, false, b, (short)0, acc[tn8], false, false);
    }
  }

  // D layout: VGPR r -> row (tm + r + 8*half), col (tn + l)
#pragma unroll
  for (int tn8 = 0; tn8 < 8; ++tn8) {
    const int col = tn8 * 16 + l;
    const float bv = bias ? bias[col] : 0.0f;
#pragma unroll
    for (int r = 0; r < 8; ++r) {
      const int row = tm + r + (half << 3);
      C[(size_t)row * N + col] = acc[tn8][r] + bv;
    }
  }
}

// --------------------------------------------------------------------------
// Fused per-graph pipeline, one 256-thread block per graph:
//   async-stage h_lin tile (64KB) + edge slice (16KB) into LDS ->
//   degree -> rsqrt -> GCN aggregation (ds_add_f32) -> ReLU ->
//   score GCNConv(HID->1) -> top-64 (top_k tie-break) ->
//   tanh-weighted mean pool -> pooled[g, HID]
// --------------------------------------------------------------------------
__global__ __launch_bounds__(256) void sag_graph_kernel(
    const int* __restrict__ src, const int* __restrict__ dst,
    const float* __restrict__ h_lin, const float* __restrict__ b1,
    const float* __restrict__ Ws, const float* __restrict__ bs,
    float* __restrict__ pooled) {
  __shared__ __align__(16) float h[NPG * HID];      // 64 KB accumulation tile
  __shared__ __align__(16) float htile[NPG * HID];  // 64 KB staged h_lin tile
  __shared__ __align__(16) int   eS[EPG];           // 8 KB staged src
  __shared__ __align__(16) int   eD[EPG];           // 8 KB staged dst
  __shared__ float lrs[NPG];
  __shared__ float zsc[NPG];
  __shared__ float sc[NPG];
  __shared__ float wgt[NPG];
  __shared__ int   keep[NPG];

  const int g     = blockIdx.x;
  const int t     = threadIdx.x;
  const int wave  = t >> 5;
  const int lane  = t & 31;
  const int ebase = g * EPG;
  const int nbase = g * NPG;

  // --- async staging: h_lin tile + edge slice (all contiguous) ---
  {
    const float* gh = h_lin + (size_t)nbase * HID;  // 16384 floats
    for (int i = t; i < (NPG * HID) / 4; i += 256)
      stage_b128(gh + i * 4, htile + i * 4);
    const int* gs = src + ebase;                    // 2048 ints = 512 x b128
    const int* gd = dst + ebase;
    for (int i = t; i < EPG / 4; i += 256) {
      stage_b128(gs + i * 4, eS + i * 4);
      stage_b128(gd + i * 4, eD + i * 4);
    }
  }
  if (t < NPG) lrs[t] = 1.0f;  // degree init (self loop)
  stage_wait();
  __syncthreads();

  // --- degree via LDS atomics ---
  for (int e = t; e < EPG; e += 256)
    atomicAdd(&lrs[eD[e] - nbase], 1.0f);
  __syncthreads();
  if (t < NPG) lrs[t] = rsqrtf(lrs[t]);
  __syncthreads();

  // --- init h = htile * rs^2 (self loop) + b1 ---
  for (int i = t; i < NPG * HID; i += 256) {
    const int v = i >> 7;
    const int f = i & (HID - 1);
    const float r = lrs[v];
    h[i] = htile[i] * r * r + b1[f];
  }
  __syncthreads();

  // --- edge aggregation: wave per edge-slice, lane covers 4 features ---
  for (int e = wave; e < EPG; e += 8) {
    const int s = eS[e] - nbase;
    const int d = eD[e] - nbase;
    const float nrm = lrs[s] * lrs[d];
    const float4* hs4 = reinterpret_cast<const float4*>(htile + s * HID);
    const float4 val = hs4[lane];
    const int f = lane << 2;
    atomicAdd(&h[d * HID + f + 0], val.x * nrm);
    atomicAdd(&h[d * HID + f + 1], val.y * nrm);
    atomicAdd(&h[d * HID + f + 2], val.z * nrm);
    atomicAdd(&h[d * HID + f + 3], val.w * nrm);
  }
  __syncthreads();

  // --- ReLU in LDS ---
  for (int i = t; i < NPG * HID; i += 256)
    h[i] = fmaxf(h[i], 0.0f);
  __syncthreads();

  // --- z[v] = dot(h[v], Ws): wave per node ---
  for (int v = wave; v < NPG; v += 8) {
    float a = 0.0f;
    for (int f = lane; f < HID; f += 32) a += h[v * HID + f] * Ws[f];
#pragma unroll
    for (int off = 16; off; off >>= 1) a += __shfl_down(a, off, 32);
    if (lane == 0) zsc[v] = a;
  }
  __syncthreads();

  // --- score: self loop + bias, then edge scatter (scalar LDS atomics) ---
  if (t < NPG) {
    const float r = lrs[t];
    sc[t] = zsc[t] * r * r + bs[0];
  }
  __syncthreads();
  for (int e = t; e < EPG; e += 256) {
    const int s = eS[e] - nbase;
    const int d = eD[e] - nbase;
    atomicAdd(&sc[d], zsc[s] * lrs[s] * lrs[d]);
  }
  __syncthreads();

  // --- top-K by rank (jax.lax.top_k tie-break: lower index wins) ---
  if (t < NPG) {
    const float sv = sc[t];
    int rank = 0;
    for (int j = 0; j < NPG; ++j) {
      const float sj = sc[j];
      rank += (sj > sv) || (sj == sv && j < t);
    }
    keep[t] = (rank < KEEPK) ? 1 : 0;
    wgt[t]  = tanhf(sv) * (1.0f / (float)KEEPK);
  }
  __syncthreads();

  // --- pooled[g, f] = mean over kept nodes of h * tanh(score) ---
  if (t < HID) {
    float a = 0.0f;
    for (int j = 0; j < NPG; ++j)
      if (keep[j]) a += h[j * HID + t] * wgt[j];
    pooled[(size_t)g * HID + t] = a;
  }
}

// --------------------------------------------------------------------------
// Launch: gemm1 (WMMA) -> fused per-graph pipeline -> gemm2 (WMMA + bias)
// Inputs: 0:x 1:edge_index 2:batch 3:W1 4:b1 5:Ws 6:bs 7:Wlin 8:blin
// --------------------------------------------------------------------------
extern "C" void kernel_launch(void* const* d_in, const int* in_sizes, int n_in,
                              void* d_out, int out_size, void* d_ws, size_t ws_size,
                              hipStream_t stream) {
  const float* x    = (const float*)d_in[0];
  const int*   ei   = (const int*)d_in[1];
  const float* W1   = (const float*)d_in[3];
  const float* b1   = (const float*)d_in[4];
  const float* Ws   = (const float*)d_in[5];
  const float* bs   = (const float*)d_in[6];
  const float* Wlin = (const float*)d_in[7];
  const float* blin = (const float*)d_in[8];
  float* out = (float*)d_out;

  const int nE = in_sizes[1] / 2;  // 1,048,576
  const int* src = ei;
  const int* dst = ei + nE;

  float* h_lin  = (float*)d_ws;                  // NTOTAL * HID
  float* pooled = h_lin + (size_t)NTOTAL * HID;  // BGRAPHS * HID

  // 1) h_lin = x @ W1   (65536x128x128): one wave per 16-row strip
  {
    const int waves  = NTOTAL / 16;  // 4096
    const int blocks = waves / 8;    // 512
    gemm_f32_wmma<<<blocks, 256, 0, stream>>>(x, W1, nullptr, h_lin,
                                              NTOTAL, HID, INF_);
  }

  // 2) fused GCN aggregation + ReLU + score + top-K + pool (block per graph)
  sag_graph_kernel<<<BGRAPHS, 256, 0, stream>>>(src, dst, h_lin, b1, Ws, bs,
                                                pooled);

  // 3) out = pooled @ Wlin + blin   (512x128x128)
  {
    const int waves  = BGRAPHS / 16;  // 32
    const int blocks = waves / 8;     // 4
    gemm_f32_wmma<<<blocks, 256, 0, stream>>>(pooled, Wlin, blin, out,
                                              BGRAPHS, HID, HID);
  }
}